// MinkResBlock_BottleNeck_51144470561090
// MI455X (gfx1250) — compile-verified
//
#include <hip/hip_runtime.h>

typedef float v2f __attribute__((ext_vector_type(2)));
typedef float v4f __attribute__((ext_vector_type(4)));
typedef float v8f __attribute__((ext_vector_type(8)));

#define N_PTS 262144
#define C_IN  256
#define C_OUT 256
#define BW    64
#define KOFF  27
#define EPS   1e-5f

// D = A(16x4 f32) * B(4x16 f32) + C(16x16 f32), full fp32 matrix pipe
__device__ __forceinline__ v8f wmma4(v2f a, v2f b, v8f c) {
  return __builtin_amdgcn_wmma_f32_16x16x4_f32(false, a, false, b, (short)0, c,
                                               false, false);
}

// ---------------------------------------------------------------------------
// Repack a [K][N] weight (batched) into WMMA B-fragment order so that a
// lane's fragment {B[k0+2hk][n], B[k0+2hk+1][n]} is one contiguous b64:
//   dst[b][ (k>>2)*2 + ((k>>1)&1) ][n][ k&1 ]
// ---------------------------------------------------------------------------
__global__ __launch_bounds__(256) void repack_w_kernel(
    const float* __restrict__ src, float* __restrict__ dst, int K, int N) {
  const long gid  = (long)blockIdx.x * 256 + threadIdx.x;
  const long mat  = (long)K * N;
  const long b    = gid / mat;
  const long off  = gid % mat;
  const int  k    = (int)(off / N);
  const int  n    = (int)(off % N);
  const long didx = b * mat + (long)((k >> 2) * 2 + ((k >> 1) & 1)) * (2 * N) +
                    2 * n + (k & 1);
  dst[didx] = src[gid];
}

// ---------------------------------------------------------------------------
// Stage 1: y1 = x @ W1a  (per-wave: 16 rows x 64 cols), + per-WG BN partials
// 256 threads = 8 waves, 128 rows per workgroup. Wp = packed W1a.
// ---------------------------------------------------------------------------
__global__ __launch_bounds__(256) void gemm1_kernel(
    const float* __restrict__ x, const float* __restrict__ Wp,
    float* __restrict__ y, float* __restrict__ part) {
  __shared__ float lds_s[8][64];
  __shared__ float lds_q[8][64];
  const int w  = threadIdx.x >> 5;
  const int l  = threadIdx.x & 31;
  const int m  = l & 15;         // M (A rows / B cols) within tile
  const int hk = l >> 4;         // lane-half selects K pair {0,1} vs {2,3}
  const long rw = (long)blockIdx.x * 128 + (long)w * 16;
  const long r  = rw + m;

  v8f acc[4] = {};
  const float* xr = x + r * C_IN + hk * 2;
#pragma unroll 4
  for (int k0 = 0; k0 < C_IN; k0 += 4) {
    v2f a = *(const v2f*)(xr + k0);
    const float* bp = Wp + (long)((k0 >> 1) + hk) * (2 * BW) + 2 * m;
#pragma unroll
    for (int t = 0; t < 4; ++t) {
      v2f b = *(const v2f*)(bp + 32 * t);
      acc[t] = wmma4(a, b, acc[t]);
    }
  }

  // store y1 tile + deterministic per-channel sum / sumsq partials
#pragma unroll
  for (int t = 0; t < 4; ++t) {
    float s = 0.f, q = 0.f;
#pragma unroll
    for (int j = 0; j < 8; ++j) {
      float v  = acc[t][j];
      long row = rw + hk * 8 + j;       // C layout: lane-half -> M +8
      y[row * BW + t * 16 + m] = v;
      s += v;
      q += v * v;
    }
    s += __shfl_xor(s, 16, 32);         // combine the two row-halves
    q += __shfl_xor(q, 16, 32);
    if (l < 16) { lds_s[w][t * 16 + l] = s; lds_q[w][t * 16 + l] = q; }
  }
  __syncthreads();
  if (threadIdx.x < 64) {
    int c = threadIdx.x;
    float s = 0.f, q = 0.f;
    for (int ww = 0; ww < 8; ++ww) { s += lds_s[ww][c]; q += lds_q[ww][c]; }
    part[(long)blockIdx.x * 128 + c]      = s;
    part[(long)blockIdx.x * 128 + 64 + c] = q;
  }
}

// ---------------------------------------------------------------------------
// BN finalize: fixed-order reduction of partials -> scale/shift (deterministic)
// ---------------------------------------------------------------------------
__global__ void bn_finalize_kernel(const float* __restrict__ part, int nParts,
                                   int C, const float* __restrict__ gamma,
                                   const float* __restrict__ beta,
                                   float* __restrict__ scale,
                                   float* __restrict__ shift) {
  int c = threadIdx.x;
  if (c >= C) return;
  float s = 0.f, q = 0.f;
  for (int p = 0; p < nParts; ++p) {
    s += part[(long)p * 2 * C + c];
    q += part[(long)p * 2 * C + C + c];
  }
  const float inv  = 1.0f / (float)N_PTS;
  const float mean = s * inv;
  const float var  = q * inv - mean * mean;   // biased var (matches jnp.var)
  const float sc   = gamma[c] * rsqrtf(var + EPS);
  scale[c] = sc;
  shift[c] = beta[c] - mean * sc;
}

// in-place h = relu(y * scale[c] + shift[c]); 4 elements / thread (b128)
__global__ __launch_bounds__(256) void bn_relu_apply_kernel(
    float* __restrict__ buf, const float* __restrict__ scale,
    const float* __restrict__ shift) {
  long i  = ((long)blockIdx.x * 256 + threadIdx.x) * 4;
  int  c0 = (int)(i & 63);                  // 4 | 64 -> never crosses channels
  v4f v = *(v4f*)(buf + i);
#pragma unroll
  for (int j = 0; j < 4; ++j) {
    float t = v[j] * scale[c0 + j] + shift[c0 + j];
    v[j] = t > 0.f ? t : 0.f;
  }
  *(v4f*)(buf + i) = v;
}

// ---------------------------------------------------------------------------
// Stage 2: y2 = sum_k h1[idx_k] @ W3[k]  (gather-GEMM; h1 is L2-resident)
// W3p = packed W3 (27 batched 64x64 mats in B-fragment order).
// ---------------------------------------------------------------------------
__global__ __launch_bounds__(256) void conv3_kernel(
    const float* __restrict__ h1, const int* __restrict__ nbr,
    const float* __restrict__ W3p, float* __restrict__ y2,
    float* __restrict__ part) {
  __shared__ float lds_s[8][64];
  __shared__ float lds_q[8][64];
  const int w  = threadIdx.x >> 5;
  const int l  = threadIdx.x & 31;
  const int m  = l & 15;
  const int hk = l >> 4;
  const long rw  = (long)blockIdx.x * 128 + (long)w * 16;
  const int iRow = (int)rw + m;

  v8f acc[4] = {};
  for (int k = 0; k < KOFF; ++k) {
    const int nb = nbr[iRow * KOFF + k];                 // gathered row
    const float* ar = h1 + (long)nb * BW + hk * 2;
    const float* wk = W3p + (long)k * BW * BW;
#pragma unroll 4
    for (int kk = 0; kk < BW; kk += 4) {
      v2f a = *(const v2f*)(ar + kk);
      const float* bp = wk + (long)((kk >> 1) + hk) * (2 * BW) + 2 * m;
#pragma unroll
      for (int t = 0; t < 4; ++t) {
        v2f b = *(const v2f*)(bp + 32 * t);
        acc[t] = wmma4(a, b, acc[t]);
      }
    }
  }

#pragma unroll
  for (int t = 0; t < 4; ++t) {
    float s = 0.f, q = 0.f;
#pragma unroll
    for (int j = 0; j < 8; ++j) {
      float v  = acc[t][j];
      long row = rw + hk * 8 + j;
      y2[row * BW + t * 16 + m] = v;
      s += v;
      q += v * v;
    }
    s += __shfl_xor(s, 16, 32);
    q += __shfl_xor(q, 16, 32);
    if (l < 16) { lds_s[w][t * 16 + l] = s; lds_q[w][t * 16 + l] = q; }
  }
  __syncthreads();
  if (threadIdx.x < 64) {
    int c = threadIdx.x;
    float s = 0.f, q = 0.f;
    for (int ww = 0; ww < 8; ++ww) { s += lds_s[ww][c]; q += lds_q[ww][c]; }
    part[(long)blockIdx.x * 128 + c]      = s;
    part[(long)blockIdx.x * 128 + 64 + c] = q;
  }
}

// ---------------------------------------------------------------------------
// Stage 3 pass 1: stats of y3 = h2 @ W1b WITHOUT storing y3 (recompute is
// cheaper than a 512MB HBM round trip). 8 waves: cg = wave&3 (64-col group),
// rg = wave>>2 (16-row group); 8 row-iterations -> 256 rows / WG.
// ---------------------------------------------------------------------------
__global__ __launch_bounds__(256) void gemm2_stats_kernel(
    const float* __restrict__ h2, const float* __restrict__ W1bp,
    float* __restrict__ part) {
  __shared__ float lds_s[8][64];
  __shared__ float lds_q[8][64];
  const int w  = threadIdx.x >> 5;
  const int l  = threadIdx.x & 31;
  const int m  = l & 15;
  const int hk = l >> 4;
  const int cg = w & 3;
  const int rg = w >> 2;

  float sums[4] = {0.f, 0.f, 0.f, 0.f};
  float sqs[4]  = {0.f, 0.f, 0.f, 0.f};
  for (int it = 0; it < 8; ++it) {
    const long rowBase = (long)blockIdx.x * 256 + it * 32 + rg * 16;
    const long r = rowBase + m;
    const float* ar = h2 + r * BW + hk * 2;
    v8f acc[4] = {};
#pragma unroll 4
    for (int k0 = 0; k0 < BW; k0 += 4) {
      v2f a = *(const v2f*)(ar + k0);
      const float* bp =
          W1bp + (long)((k0 >> 1) + hk) * (2 * C_OUT) + cg * 128 + 2 * m;
#pragma unroll
      for (int t = 0; t < 4; ++t) {
        v2f b = *(const v2f*)(bp + 32 * t);
        acc[t] = wmma4(a, b, acc[t]);
      }
    }
#pragma unroll
    for (int t = 0; t < 4; ++t) {
      float s = 0.f, q = 0.f;
#pragma unroll
      for (int j = 0; j < 8; ++j) { float v = acc[t][j]; s += v; q += v * v; }
      sums[t] += s;
      sqs[t]  += q;
    }
  }
#pragma unroll
  for (int t = 0; t < 4; ++t) {
    float s = sums[t] + __shfl_xor(sums[t], 16, 32);
    float q = sqs[t]  + __shfl_xor(sqs[t], 16, 32);
    if (l < 16) { lds_s[w][t * 16 + l] = s; lds_q[w][t * 16 + l] = q; }
  }
  __syncthreads();
  {
    int c  = threadIdx.x;                 // global channel 0..255
    int wa = c >> 6;                      // wave with cg == c>>6
    int cl = c & 63;
    float s = lds_s[wa][cl] + lds_s[wa + 4][cl];
    float q = lds_q[wa][cl] + lds_q[wa + 4][cl];
    part[(long)blockIdx.x * 512 + c]       = s;
    part[(long)blockIdx.x * 512 + 256 + c] = q;
  }
}

// Stage 3 pass 2: out = relu(bn3(h2 @ W1b) + x)   (32 rows x 256 cols per WG)
__global__ __launch_bounds__(256) void gemm2_out_kernel(
    const float* __restrict__ h2, const float* __restrict__ W1bp,
    const float* __restrict__ x, const float* __restrict__ scale,
    const float* __restrict__ shift, float* __restrict__ out) {
  const int w  = threadIdx.x >> 5;
  const int l  = threadIdx.x & 31;
  const int m  = l & 15;
  const int hk = l >> 4;
  const int cg = w & 3;
  const int rg = w >> 2;
  const long rowBase = (long)blockIdx.x * 32 + rg * 16;
  const long r = rowBase + m;

  const float* ar = h2 + r * BW + hk * 2;
  v8f acc[4] = {};
#pragma unroll 4
  for (int k0 = 0; k0 < BW; k0 += 4) {
    v2f a = *(const v2f*)(ar + k0);
    const float* bp =
        W1bp + (long)((k0 >> 1) + hk) * (2 * C_OUT) + cg * 128 + 2 * m;
#pragma unroll
    for (int t = 0; t < 4; ++t) {
      v2f b = *(const v2f*)(bp + 32 * t);
      acc[t] = wmma4(a, b, acc[t]);
    }
  }
#pragma unroll
  for (int t = 0; t < 4; ++t) {
    const int col  = cg * 64 + t * 16 + m;
    const float sc = scale[col];
    const float sh = shift[col];
#pragma unroll
    for (int j = 0; j < 8; ++j) {
      const long row = rowBase + hk * 8 + j;
      float v = acc[t][j] * sc + sh + x[row * C_OUT + col];
      out[row * C_OUT + col] = v > 0.f ? v : 0.f;
    }
  }
}

// ---------------------------------------------------------------------------
extern "C" void kernel_launch(void* const* d_in, const int* in_sizes, int n_in,
                              void* d_out, int out_size, void* d_ws,
                              size_t ws_size, hipStream_t stream) {
  const float* x   = (const float*)d_in[0];
  const int*   nbr = (const int*)d_in[1];
  const float* W1a = (const float*)d_in[2];
  const float* g1a = (const float*)d_in[3];
  const float* b1a = (const float*)d_in[4];
  const float* W3  = (const float*)d_in[5];
  const float* g3  = (const float*)d_in[6];
  const float* b3  = (const float*)d_in[7];
  const float* W1b = (const float*)d_in[8];
  const float* g1b = (const float*)d_in[9];
  const float* b1b = (const float*)d_in[10];
  float* out = (float*)d_out;

  float* ws     = (float*)d_ws;
  float* buf1   = ws;                                   // y1 -> h1 (in place)
  float* buf2   = buf1 + (size_t)N_PTS * BW;            // y2 -> h2 (in place)
  float* part12 = buf2 + (size_t)N_PTS * BW;            // 2048 * 128
  float* part3  = part12 + (size_t)2048 * 128;          // 1024 * 512
  float* ss     = part3 + (size_t)1024 * 512;           // scale/shift x3 (768)
  float* W1aP   = ss + 768;                             // 256*64
  float* W3P    = W1aP + (size_t)C_IN * BW;             // 27*64*64
  float* W1bP   = W3P + (size_t)KOFF * BW * BW;         // 64*256

  const int nWG1 = N_PTS / 128;   // 2048
  const int nWG3 = N_PTS / 256;   // 1024

  // Repack weights into WMMA B-fragment order (one b64 per fragment)
  repack_w_kernel<<<(C_IN * BW) / 256, 256, 0, stream>>>(W1a, W1aP, C_IN, BW);
  repack_w_kernel<<<(KOFF * BW * BW) / 256, 256, 0, stream>>>(W3, W3P, BW, BW);
  repack_w_kernel<<<(BW * C_OUT) / 256, 256, 0, stream>>>(W1b, W1bP, BW,
                                                          C_OUT);

  // Stage 1
  gemm1_kernel<<<nWG1, 256, 0, stream>>>(x, W1aP, buf1, part12);
  bn_finalize_kernel<<<1, 64, 0, stream>>>(part12, nWG1, 64, g1a, b1a,
                                           ss + 0, ss + 64);
  bn_relu_apply_kernel<<<(N_PTS * BW) / 1024, 256, 0, stream>>>(buf1, ss + 0,
                                                                ss + 64);
  // Stage 2
  conv3_kernel<<<nWG1, 256, 0, stream>>>(buf1, nbr, W3P, buf2, part12);
  bn_finalize_kernel<<<1, 64, 0, stream>>>(part12, nWG1, 64, g3, b3,
                                           ss + 128, ss + 192);
  bn_relu_apply_kernel<<<(N_PTS * BW) / 1024, 256, 0, stream>>>(buf2, ss + 128,
                                                                ss + 192);
  // Stage 3
  gemm2_stats_kernel<<<nWG3, 256, 0, stream>>>(buf2, W1bP, part3);
  bn_finalize_kernel<<<1, 256, 0, stream>>>(part3, nWG3, 256, g1b, b1b,
                                            ss + 256, ss + 512);
  gemm2_out_kernel<<<N_PTS / 32, 256, 0, stream>>>(buf2, W1bP, x, ss + 256,
                                                   ss + 512, out);
}